// GNN_82592221102580
// MI455X (gfx1250) — compile-verified
//
#include <hip/hip_runtime.h>
#include <math.h>

// CDNA5 wave32 WMMA fragment types for V_WMMA_F32_16X16X4_F32:
//  A 16x4 f32  -> 2 VGPRs/lane, B 4x16 f32 -> 2 VGPRs/lane, C/D 16x16 f32 -> 8 VGPRs/lane
typedef __attribute__((ext_vector_type(2))) float v2f;
typedef __attribute__((ext_vector_type(8))) float v8f;

#define LS 68  // LDS row stride in floats: 64 + 4 pad (16B-aligned rows, bank spread)

__global__ __launch_bounds__(256) void gnn_fused_kernel(
    const float* __restrict__ X,   // [4096, 64, 64]
    const float* __restrict__ W,   // [64, 64]
    const float* __restrict__ a2,  // [64]
    const float* __restrict__ b2,  // [64]
    float* __restrict__ out)       // [4096, 64, 64]
{
    __shared__ float sX[64 * LS];   // X -> Xn (in place) -> W (reused)
    __shared__ float sA[64 * LS];   // A = softmax(X^T) [w][h] -> out staging (reused)
    __shared__ float sN[64 * LS];   // node = A @ Xn
    __shared__ float sRed[512];     // reduction scratch (sum & sumsq)
    __shared__ float sStat0[64];    // colmax   / row mean
    __shared__ float sStat1[64];    // 1/colsum / 1/(std+eps)
    __shared__ float sG[64];
    __shared__ float sB[64];

    const int tid = threadIdx.x;
    const size_t batch = blockIdx.x;
    const float* Xb = X + batch * 4096;
    float* Ob = out + batch * 4096;

    // ---- issue W fetch EARLY into registers (hidden behind softmax/LN/GEMM1) ----
    const float4* Wv = reinterpret_cast<const float4*>(W);
    float4 w0 = Wv[tid];
    float4 w1 = Wv[tid + 256];
    float4 w2 = Wv[tid + 512];
    float4 w3 = Wv[tid + 768];

    // ---------------- stage X tile + affine params ----------------
    for (int i = tid; i < 1024; i += 256) {
        float4 v = reinterpret_cast<const float4*>(Xb)[i];
        *reinterpret_cast<float4*>(&sX[(i >> 4) * LS + ((i & 15) << 2)]) = v;
    }
    if (tid < 64) { sG[tid] = a2[tid]; sB[tid] = b2[tid]; }
    __syncthreads();

    // ---------------- softmax over h per column c: A[c][j] ----------------
    {
        const int c  = tid & 63;       // column of X (w index)
        const int sg = tid >> 6;       // 0..3 -> 16-row segment
        const int j0 = sg << 4;

        float m = -3.402823466e+38f;
        #pragma unroll
        for (int j = 0; j < 16; ++j) m = fmaxf(m, sX[(j0 + j) * LS + c]);
        sRed[sg * 64 + c] = m;
        __syncthreads();
        if (tid < 64) {
            sStat0[tid] = fmaxf(fmaxf(sRed[tid], sRed[64 + tid]),
                                fmaxf(sRed[128 + tid], sRed[192 + tid]));
        }
        __syncthreads();

        const float cm = sStat0[c];
        float ssum = 0.f;
        #pragma unroll
        for (int j = 0; j < 16; ++j) {
            float e = __expf(sX[(j0 + j) * LS + c] - cm);
            sA[c * LS + (j0 + j)] = e;   // transposed store: A is [w][h]
            ssum += e;
        }
        sRed[sg * 64 + c] = ssum;
        __syncthreads();
        if (tid < 64) {
            float t = sRed[tid] + sRed[64 + tid] + sRed[128 + tid] + sRed[192 + tid];
            sStat1[tid] = 1.0f / t;
        }
        __syncthreads();

        const float ci = sStat1[c];
        #pragma unroll
        for (int j = 0; j < 16; ++j) sA[c * LS + (j0 + j)] *= ci;
    }

    // ---------------- row LayerNorm (ddof=1) of X, in place ----------------
    {
        const int r  = tid >> 2;       // row (h index)
        const int q  = tid & 3;
        const int c0 = q << 4;

        float s1 = 0.f, s2 = 0.f;
        #pragma unroll
        for (int c = 0; c < 16; ++c) {
            float v = sX[r * LS + c0 + c];
            s1 += v; s2 += v * v;
        }
        sRed[q * 64 + r]       = s1;
        sRed[256 + q * 64 + r] = s2;
        __syncthreads();
        if (tid < 64) {
            float t1 = sRed[tid] + sRed[64 + tid] + sRed[128 + tid] + sRed[192 + tid];
            float t2 = sRed[256 + tid] + sRed[320 + tid] + sRed[384 + tid] + sRed[448 + tid];
            float mu  = t1 * (1.0f / 64.0f);
            float var = (t2 - 64.0f * mu * mu) * (1.0f / 63.0f);  // unbiased (ddof=1)
            float sd  = sqrtf(fmaxf(var, 0.0f));
            sStat0[tid] = mu;
            sStat1[tid] = 1.0f / (sd + 1e-6f);
        }
        __syncthreads();

        const float mu  = sStat0[r];
        const float inv = sStat1[r];
        #pragma unroll
        for (int c = 0; c < 16; ++c) {
            int cc = c0 + c;
            float v = (sX[r * LS + cc] - mu) * inv;
            sX[r * LS + cc] = sG[cc] * v + sB[cc];
        }
    }
    __syncthreads();

    // ---------------- WMMA tiling: each wave -> 16-row strip, 2 col tiles ---
    const int lane = tid & 31;
    const int wv   = tid >> 5;            // 0..7
    const int tm   = wv >> 1;             // 0..3  (M tile)
    const int tn0  = (wv & 1) << 1;       // 0 or 2 (first N tile of pair)
    const int mr   = lane & 15;           // m (A) or n (B/D) within tile
    const int kh   = lane >> 4;           // 0/1: K-half of frag, row-half of D
    const int k0   = kh << 1;             // 0 or 2

    // ---------------- GEMM1: node = A[w,h] @ Xn[h,w] ----------------
    {
        v8f acc0 = {}; v8f acc1 = {};
        const int arow = (tm * 16 + mr) * LS;
        #pragma unroll
        for (int kb = 0; kb < 16; ++kb) {
            const int kk = (kb << 2) + k0;
            v2f a;
            a.x = sA[arow + kk];
            a.y = sA[arow + kk + 1];
            v2f b0, b1;
            b0.x = sX[kk * LS + tn0 * 16 + mr];
            b0.y = sX[(kk + 1) * LS + tn0 * 16 + mr];
            b1.x = sX[kk * LS + (tn0 + 1) * 16 + mr];
            b1.y = sX[(kk + 1) * LS + (tn0 + 1) * 16 + mr];
            acc0 = __builtin_amdgcn_wmma_f32_16x16x4_f32(false, a, false, b0,
                                                         (short)0, acc0, false, false);
            acc1 = __builtin_amdgcn_wmma_f32_16x16x4_f32(false, a, false, b1,
                                                         (short)0, acc1, false, false);
        }
        #pragma unroll
        for (int r = 0; r < 8; ++r) {
            int row = tm * 16 + (kh << 3) + r;          // D layout: VGPR r <-> M = r + 8*kh
            sN[row * LS + tn0 * 16 + mr]       = acc0[r];
            sN[row * LS + (tn0 + 1) * 16 + mr] = acc1[r];
        }
    }
    __syncthreads();

    // ---- dump preloaded W registers into the dead Xn buffer (no global fetch here) ----
    {
        int i0 = tid;
        *reinterpret_cast<float4*>(&sX[(i0 >> 4) * LS + ((i0 & 15) << 2)]) = w0; i0 += 256;
        *reinterpret_cast<float4*>(&sX[(i0 >> 4) * LS + ((i0 & 15) << 2)]) = w1; i0 += 256;
        *reinterpret_cast<float4*>(&sX[(i0 >> 4) * LS + ((i0 & 15) << 2)]) = w2; i0 += 256;
        *reinterpret_cast<float4*>(&sX[(i0 >> 4) * LS + ((i0 & 15) << 2)]) = w3;
    }
    __syncthreads();

    // ---------------- GEMM2: out = relu(node @ W) ----------------
    {
        v8f acc0 = {}; v8f acc1 = {};
        const int arow = (tm * 16 + mr) * LS;
        #pragma unroll
        for (int kb = 0; kb < 16; ++kb) {
            const int kk = (kb << 2) + k0;
            v2f a;
            a.x = sN[arow + kk];
            a.y = sN[arow + kk + 1];
            v2f b0, b1;
            b0.x = sX[kk * LS + tn0 * 16 + mr];
            b0.y = sX[(kk + 1) * LS + tn0 * 16 + mr];
            b1.x = sX[kk * LS + (tn0 + 1) * 16 + mr];
            b1.y = sX[(kk + 1) * LS + (tn0 + 1) * 16 + mr];
            acc0 = __builtin_amdgcn_wmma_f32_16x16x4_f32(false, a, false, b0,
                                                         (short)0, acc0, false, false);
            acc1 = __builtin_amdgcn_wmma_f32_16x16x4_f32(false, a, false, b1,
                                                         (short)0, acc1, false, false);
        }
        // relu + stage into dead sA buffer for coalesced b128 global stores
        #pragma unroll
        for (int r = 0; r < 8; ++r) {
            int row = tm * 16 + (kh << 3) + r;
            sA[row * LS + tn0 * 16 + mr]       = fmaxf(acc0[r], 0.0f);
            sA[row * LS + (tn0 + 1) * 16 + mr] = fmaxf(acc1[r], 0.0f);
        }
    }
    __syncthreads();

    // ---------------- coalesced float4 writeback of the 16 KB tile ----------------
    for (int i = tid; i < 1024; i += 256) {
        float4 v = *reinterpret_cast<const float4*>(&sA[(i >> 4) * LS + ((i & 15) << 2)]);
        reinterpret_cast<float4*>(Ob)[i] = v;
    }
}

extern "C" void kernel_launch(void* const* d_in, const int* in_sizes, int n_in,
                              void* d_out, int out_size, void* d_ws, size_t ws_size,
                              hipStream_t stream) {
    (void)n_in; (void)out_size; (void)d_ws; (void)ws_size;
    const float* X  = (const float*)d_in[0];
    const float* W  = (const float*)d_in[1];
    const float* a2 = (const float*)d_in[2];
    const float* b2 = (const float*)d_in[3];
    float* out = (float*)d_out;

    const int batches = in_sizes[0] / 4096;  // n*c = 4096 batches of 64x64
    gnn_fused_kernel<<<dim3(batches), dim3(256), 0, stream>>>(X, W, a2, b2, out);
}